// ModalFusionModule_64656437674065
// MI455X (gfx1250) — compile-verified
//
#include <hip/hip_runtime.h>
#include <math.h>
#include <stdint.h>

#define HW   4096
#define CC   256
#define BB   8
#define NP   30
#define NI   40
#define W_I  0.7f
#define C1C  1.5f
#define C2C  1.5f
#define LAMB 0.3f
#define EPSF 1e-8f
#define TPB  4      // px tiles per block in the GEMM pipeline

typedef __attribute__((ext_vector_type(2))) float v2f;
typedef __attribute__((ext_vector_type(8))) float v8f;
typedef __attribute__((ext_vector_type(4))) unsigned int u32x4;
typedef __attribute__((ext_vector_type(4))) int i32x4;
typedef __attribute__((ext_vector_type(8))) int i32x8;

// ---------------------------------------------------------------------------
// threefry2x32 (exact JAX semantics)
// ---------------------------------------------------------------------------
__device__ __forceinline__ uint32_t rotl32(uint32_t x, int n) {
  return (x << n) | (x >> (32 - n));
}

__device__ __forceinline__ void threefry2x32(uint32_t k0, uint32_t k1,
                                             uint32_t c0, uint32_t c1,
                                             uint32_t& o0, uint32_t& o1) {
  uint32_t ks0 = k0, ks1 = k1, ks2 = k0 ^ k1 ^ 0x1BD11BDAu;
  uint32_t x0 = c0 + ks0, x1 = c1 + ks1;
#define TF_R(r) { x0 += x1; x1 = rotl32(x1, (r)); x1 ^= x0; }
  TF_R(13) TF_R(15) TF_R(26) TF_R(6)  x0 += ks1; x1 += ks2 + 1u;
  TF_R(17) TF_R(29) TF_R(16) TF_R(24) x0 += ks2; x1 += ks0 + 2u;
  TF_R(13) TF_R(15) TF_R(26) TF_R(6)  x0 += ks0; x1 += ks1 + 3u;
  TF_R(17) TF_R(29) TF_R(16) TF_R(24) x0 += ks1; x1 += ks2 + 4u;
  TF_R(13) TF_R(15) TF_R(26) TF_R(6)  x0 += ks2; x1 += ks0 + 5u;
#undef TF_R
  o0 = x0; o1 = x1;
}

__device__ __forceinline__ float jax_uniform(uint32_t k0, uint32_t k1,
                                             uint32_t m, uint32_t half,
                                             float lo, float hi) {
  uint32_t c0 = (m < half) ? m : (m - half);
  uint32_t c1 = c0 + half;
  uint32_t y0, y1;
  threefry2x32(k0, k1, c0, c1, y0, y1);
  uint32_t bits = (m < half) ? y0 : y1;
  float u = __uint_as_float(0x3f800000u | (bits >> 9)) - 1.0f;
  return fmaxf(lo, u * (hi - lo) + lo);
}

// ---------------------------------------------------------------------------
// Kernel 1: cosine-similarity maps  rgb_sim/dep_sim [B, HW]
// ---------------------------------------------------------------------------
__global__ void sim_kernel(const float* __restrict__ f_rgb,
                           const float* __restrict__ f_depth,
                           const float* __restrict__ proto_rgb,
                           const float* __restrict__ proto_dep,
                           float* __restrict__ rgb_sim,
                           float* __restrict__ dep_sim) {
  const int b   = blockIdx.x >> 4;
  const int px  = ((blockIdx.x & 15) << 8) + threadIdx.x;
  __shared__ float sr[256], sd[256];
  float pr = proto_rgb[b * CC + threadIdx.x];
  float pd = proto_dep[b * CC + threadIdx.x];
  sr[threadIdx.x] = pr * pr;
  sd[threadIdx.x] = pd * pd;
  __syncthreads();
  for (int s = 128; s > 0; s >>= 1) {
    if (threadIdx.x < s) { sr[threadIdx.x] += sr[threadIdx.x + s];
                           sd[threadIdx.x] += sd[threadIdx.x + s]; }
    __syncthreads();
  }
  const float pnr = fmaxf(sqrtf(sr[0]), 1e-12f);
  const float pnd = fmaxf(sqrtf(sd[0]), 1e-12f);

  const float* fr  = f_rgb   + (size_t)b * CC * HW + px;
  const float* fd  = f_depth + (size_t)b * CC * HW + px;
  const float* prp = proto_rgb + b * CC;
  const float* pdp = proto_dep + b * CC;
  float dr = 0.f, nr = 0.f, dd = 0.f, nd = 0.f;
  for (int c = 0; c < CC; ++c) {
    float a = fr[(size_t)c * HW];
    float g = fd[(size_t)c * HW];
    dr += a * prp[c];  nr += a * a;
    dd += g * pdp[c];  nd += g * g;
  }
  const float fnr = fmaxf(sqrtf(nr), 1e-12f);
  const float fnd = fmaxf(sqrtf(nd), 1e-12f);
  rgb_sim[b * HW + px] = dr / (fnr * pnr);
  dep_sim[b * HW + px] = dd / (fnd * pnd);
}

// ---------------------------------------------------------------------------
// Kernel 2: conv_w [o,c] -> Wt2 pair-interleaved: Wt2[(c/2)*512 + o*2 + (c&1)]
// so the WMMA A operand (c, c+1 at fixed o) is one b64 load.
// ---------------------------------------------------------------------------
__global__ void transpose_kernel(const float* __restrict__ w,
                                 float* __restrict__ wt2) {
  int idx = blockIdx.x * 256 + threadIdx.x;
  int o = idx >> 8, c = idx & 255;
  wt2[((c >> 1) << 9) + (o << 1) + (c & 1)] = w[idx];
}

// ---------------------------------------------------------------------------
// Kernel 3: pixel statistics for closed-form PSO fitness.
// ---------------------------------------------------------------------------
__global__ void stats_kernel(const float* __restrict__ rgb_sim,
                             const float* __restrict__ dep_sim,
                             const float* __restrict__ query_mask,
                             const float* __restrict__ r_depth,
                             float* __restrict__ stats) {
  const int tid = threadIdx.x;
  float acc[22];
#pragma unroll
  for (int q = 0; q < 22; ++q) acc[q] = 0.f;

  for (int i = 0; i < 16; ++i) {
    const int px = tid + (i << 8);
    float r = 0.f, d = 0.f, u = 0.f, rd = 0.f;
#pragma unroll
    for (int b = 0; b < BB; ++b) {
      r  += rgb_sim[b * HW + px];
      d  += dep_sim[b * HW + px];
      u  += query_mask[b * HW + px];
      rd += r_depth[b * HW + px];
    }
    r *= 0.125f; d *= 0.125f; u *= 0.125f; rd *= 0.125f;
    const float A = u * r, Bv = u * d, Cv = (1.f - u) * r, Dv = (1.f - u) * d;
    acc[0]  += u;        acc[1]  += u * u;
    acc[2]  += A;        acc[3]  += Bv;      acc[4]  += Cv;      acc[5]  += Dv;
    acc[6]  += A * u;    acc[7]  += Bv * u;  acc[8]  += Cv * u;  acc[9]  += Dv * u;
    acc[10] += A * A;    acc[11] += A * Bv;  acc[12] += A * Cv;  acc[13] += A * Dv;
    acc[14] += Bv * Bv;  acc[15] += Bv * Cv; acc[16] += Bv * Dv;
    acc[17] += Cv * Cv;  acc[18] += Cv * Dv; acc[19] += Dv * Dv;
    acc[20] += (1.f - rd) * u;
    acc[21] += (1.f - rd) * (1.f - u);
  }

  __shared__ float s[256];
  __shared__ float out[22];
  for (int q = 0; q < 22; ++q) {
    s[tid] = acc[q];
    __syncthreads();
    for (int st = 128; st > 0; st >>= 1) {
      if (tid < st) s[tid] += s[tid + st];
      __syncthreads();
    }
    if (tid == 0) out[q] = s[0];
    __syncthreads();
  }

  if (tid == 0) {
    const float N = 4096.f;
    const float su = out[0], suu = out[1];
    const float mbar = su / N;
    stats[0] = sqrtf(fmaxf(suu - su * su / N, 0.f)) + EPSF;
    stats[1] = out[2]; stats[2] = out[3]; stats[3] = out[4]; stats[4] = out[5];
    stats[5] = out[6] - mbar * out[2];
    stats[6] = out[7] - mbar * out[3];
    stats[7] = out[8] - mbar * out[4];
    stats[8] = out[9] - mbar * out[5];
#pragma unroll
    for (int q = 0; q < 10; ++q) stats[9 + q] = out[10 + q];
    stats[19] = out[20] / N;
    stats[20] = out[21] / N;
  }
}

// ---------------------------------------------------------------------------
// Kernel 4: PSO, one wave32, one lane per particle, exact JAX threefry RNG.
// ---------------------------------------------------------------------------
struct PsoStats {
  float mn_norm;
  float S[4];
  float M[4];
  float G[10];   // AA AB AC AD BB BC BD CC CD DD
  float pen_fg, pen_bg;
};

__device__ __forceinline__ float pso_fitness(const float p[6], const PsoStats& st) {
  const float e0 = expf(p[0]), e1 = expf(p[1]), e2 = expf(p[2]), e3 = expf(p[3]);
  const float fgs = e0 + e1 + EPSF, bgs = e2 + e3 + EPSF;
  const float c0 = e0 / fgs, c1 = e1 / fgs, c2 = e2 / bgs, c3 = e3 / bgs;
  const float num = c0 * st.M[0] + c1 * st.M[1] + c2 * st.M[2] + c3 * st.M[3];
  const float q =
      c0 * c0 * st.G[0] + c1 * c1 * st.G[4] + c2 * c2 * st.G[7] + c3 * c3 * st.G[9] +
      2.f * (c0 * c1 * st.G[1] + c0 * c2 * st.G[2] + c0 * c3 * st.G[3] +
             c1 * c2 * st.G[5] + c1 * c3 * st.G[6] + c2 * c3 * st.G[8]);
  const float fm = (c0 * st.S[0] + c1 * st.S[1] + c2 * st.S[2] + c3 * st.S[3]) / 4096.f;
  const float fn_norm = sqrtf(fmaxf(q - 4096.f * fm * fm, 0.f));
  const float ncc = num / (fn_norm * st.mn_norm + EPSF);
  return ncc - LAMB * (c1 * st.pen_fg + c3 * st.pen_bg);
}

__global__ void pso_kernel(const float* __restrict__ stats,
                           float* __restrict__ pso_out) {
  const int lane = threadIdx.x;
  const bool active = lane < NP;

  PsoStats st;
  st.mn_norm = stats[0];
#pragma unroll
  for (int i = 0; i < 4; ++i)  st.S[i] = stats[1 + i];
#pragma unroll
  for (int i = 0; i < 4; ++i)  st.M[i] = stats[5 + i];
#pragma unroll
  for (int i = 0; i < 10; ++i) st.G[i] = stats[9 + i];
  st.pen_fg = stats[19];
  st.pen_bg = stats[20];

  uint32_t y00, y10, y01, y11, y02, y12, y03, y13;
  threefry2x32(0u, 42u, 0u, 4u, y00, y10);
  threefry2x32(0u, 42u, 1u, 5u, y01, y11);
  threefry2x32(0u, 42u, 2u, 6u, y02, y12);
  threefry2x32(0u, 42u, 3u, 7u, y03, y13);
  const uint32_t kp0 = y00, kp1 = y01;
  const uint32_t kv0 = y02, kv1 = y03;
  const uint32_t k10 = y10, k11 = y11;
  const uint32_t k20 = y12, k21 = y13;

  float pos[6], vel[6], pb[6], gb[6];
#pragma unroll
  for (int j = 0; j < 6; ++j) {
    const uint32_t m = (uint32_t)(lane * 6 + j);
    pos[j] = active ? jax_uniform(kp0, kp1, m, 90u, -2.0f, 2.0f) : 0.f;
    vel[j] = active ? jax_uniform(kv0, kv1, m, 90u, -0.5f, 0.5f) : 0.f;
    pb[j] = pos[j];
  }
  float pb_fit = active ? pso_fitness(pos, st) : -INFINITY;

  float bf = pb_fit; int bi = lane;
  for (int off = 16; off >= 1; off >>= 1) {
    float of = __shfl_xor(bf, off, 32);
    int   oi = __shfl_xor(bi, off, 32);
    if (of > bf || (of == bf && oi < bi)) { bf = of; bi = oi; }
  }
  float gb_fit = bf;
#pragma unroll
  for (int j = 0; j < 6; ++j) gb[j] = __shfl(pb[j], bi, 32);

  for (int it = 0; it < NI; ++it) {
#pragma unroll
    for (int j = 0; j < 6; ++j) {
      float r1 = 0.f, r2 = 0.f;
      if (active) {
        const uint32_t m = (uint32_t)(it * 180 + lane * 6 + j);
        r1 = jax_uniform(k10, k11, m, 3600u, 0.f, 1.f);
        r2 = jax_uniform(k20, k21, m, 3600u, 0.f, 1.f);
      }
      vel[j] = W_I * vel[j] + C1C * r1 * (pb[j] - pos[j]) + C2C * r2 * (gb[j] - pos[j]);
      pos[j] = fminf(fmaxf(pos[j] + vel[j], -3.0f), 3.0f);
    }
    const float fit = active ? pso_fitness(pos, st) : -INFINITY;
    if (fit > pb_fit) {
      pb_fit = fit;
#pragma unroll
      for (int j = 0; j < 6; ++j) pb[j] = pos[j];
    }
    bf = pb_fit; bi = lane;
    for (int off = 16; off >= 1; off >>= 1) {
      float of = __shfl_xor(bf, off, 32);
      int   oi = __shfl_xor(bi, off, 32);
      if (of > bf || (of == bf && oi < bi)) { bf = of; bi = oi; }
    }
    if (bf > gb_fit) {
      gb_fit = bf;
#pragma unroll
      for (int j = 0; j < 6; ++j) gb[j] = __shfl(pb[j], bi, 32);
    }
  }

  if (lane == 0) {
    const float e0 = expf(gb[0]), e1 = expf(gb[1]), e2 = expf(gb[2]), e3 = expf(gb[3]);
    const float fgs = e0 + e1 + EPSF, bgs = e2 + e3 + EPSF;
    pso_out[0] = e0 / fgs;
    pso_out[1] = e1 / fgs;
    pso_out[2] = e2 / bgs;
    pso_out[3] = e3 / bgs;
  }
}

// ---------------------------------------------------------------------------
// TDM: 2-D tensor_load_to_lds of one 16(px) x 256(c) fp32 tile (stride 4096).
// D# group0: count=1, lds_addr, global_addr(57b), type=2.
// D# group1: data_size=4B, tensor_dim0=4096, tensor_dim1=256,
//            tile_dim0=16, tile_dim1=256, tensor_dim0_stride=4096.
// 6-arg builtin on this toolchain (clang-23 / therock headers).
// ---------------------------------------------------------------------------
__device__ __forceinline__ void tdm_load_tile_2d(uint64_t gaddr, uint32_t lds_addr) {
  u32x4 g0;
  g0.x = 1u;                                             // count=1 (valid user D#)
  g0.y = lds_addr;                                       // lds_addr [63:32]
  g0.z = (unsigned)(gaddr & 0xFFFFFFFFull);              // global_addr [95:64]
  g0.w = (unsigned)((gaddr >> 32) & 0x1FFFFFFull)        // global_addr [120:96]
         | (2u << 30);                                   // type=2 ("image")
  i32x8 g1;
  g1[0] = (int)0x00020000;   // data_size=4B (bits 17:16 = 2)
  g1[1] = (int)0x10000000;   // tensor_dim0 = 4096 (bits 63:48)
  g1[2] = (int)0x01000000;   // tensor_dim1 = 256 (bits 111:80 -> low16 in [31:16])
  g1[3] = (int)0x00100000;   // tile_dim0 = 16 (bits 127:112)
  g1[4] = 256;               // tile_dim1 = 256 (bits 143:128)
  g1[5] = 4096;              // tensor_dim0_stride[31:0] (bits 191:160)
  g1[6] = 0;                 // stride hi / tensor_dim1_stride lo
  g1[7] = 0;
  i32x4 z4 = {0, 0, 0, 0};
  i32x8 z8 = {0, 0, 0, 0, 0, 0, 0, 0};
  __builtin_amdgcn_tensor_load_to_lds(g0, g1, z4, z4, z8, 0);
}

// ---------------------------------------------------------------------------
// Kernel 5: pipelined WMMA GEMM.
//   grid (64, 8), 256 threads (8 waves). Each block: 4 px-tiles of one batch.
//   Per tile: TDM stages raw rgb/dep tiles -> LDS; a conversion pass builds the
//   proxy-fused operand F (pair-interleaved); 8 waves x 2 o-tiles x 64 WMMA
//   steps consume F; TDM for tile t+1 overlaps the WMMA sweep of tile t.
// ---------------------------------------------------------------------------
__global__ void __launch_bounds__(256)
gemm_kernel(const float* __restrict__ f_rgb,
            const float* __restrict__ f_depth,
            const float* __restrict__ rgb_sim,
            const float* __restrict__ dep_sim,
            const float* __restrict__ Wt2,
            const float* __restrict__ pso_out,
            const float* __restrict__ bn_gamma,
            const float* __restrict__ bn_beta,
            const float* __restrict__ bn_mean,
            const float* __restrict__ bn_var,
            float* __restrict__ out) {
  __shared__ __align__(64) float lds_raw[8192];   // [0:4096) rgb, [4096:8192) dep
  __shared__ __align__(64) float lds_F[4096];     // pair-interleaved fused operand

  const int b    = blockIdx.y;
  const int tid  = threadIdx.x;
  const int wave = tid >> 5;
  const int lane = tid & 31;
  const int half = lane >> 4;
  const int l    = lane & 15;
  const int tile0 = blockIdx.x * TPB;

  const uint64_t frb = (uint64_t)(uintptr_t)(f_rgb   + (size_t)b * CC * HW);
  const uint64_t fdb = (uint64_t)(uintptr_t)(f_depth + (size_t)b * CC * HW);
  const uint32_t raw_r_lds = (uint32_t)(uintptr_t)&lds_raw[0];
  const uint32_t raw_d_lds = (uint32_t)(uintptr_t)&lds_raw[4096];

  const float rgb_fg = pso_out[0], dep_fg = pso_out[1];
  const float rgb_bg = pso_out[2], dep_bg = pso_out[3];

  // BN constants for this wave's 2 o-tiles (per-lane: 8 rows each)
  float binv[2][8], badd[2][8];
#pragma unroll
  for (int ot = 0; ot < 2; ++ot) {
    const int ob = (((wave << 1) + ot) << 4) + (half << 3);
#pragma unroll
    for (int v = 0; v < 8; ++v) {
      const int o = ob + v;
      const float inv = bn_gamma[o] * rsqrtf(bn_var[o] + 1e-5f);
      binv[ot][v] = inv;
      badd[ot][v] = bn_beta[o] - bn_mean[o] * inv;
    }
  }

  // prologue: stage tile 0
  if (wave == 0) {
    const uint64_t pxb = (uint64_t)(tile0 << 4) * 4u;
    tdm_load_tile_2d(frb + pxb, raw_r_lds);
    tdm_load_tile_2d(fdb + pxb, raw_d_lds);
  }

  for (int t = 0; t < TPB; ++t) {
    const int px0 = (tile0 + t) << 4;

    if (wave == 0) __builtin_amdgcn_s_wait_tensorcnt((short)0);
    __syncthreads();

    // conversion pass: F[(c/2)*32 + l*2 + (c&1)] = wr*rgb + wd*dep
    {
      const int  myl  = tid & 15;
      const int  mypx = px0 + myl;
      const float ssum  = rgb_sim[b * HW + mypx] + dep_sim[b * HW + mypx];
      const float proxy = 1.0f / (1.0f + expf(-3.0f * ssum));
      const float wr = proxy * rgb_fg + (1.0f - proxy) * rgb_bg;
      const float wd = proxy * dep_fg + (1.0f - proxy) * dep_bg;
#pragma unroll
      for (int i = 0; i < 8; ++i) {
        const int pidx = (i << 8) + tid;   // pair index space: 0..2047
        const int cp   = pidx >> 4;        // 0..127
        const int c0   = cp << 1;
        const float fr0 = lds_raw[c0 * 16 + myl];
        const float fr1 = lds_raw[(c0 + 1) * 16 + myl];
        const float fd0 = lds_raw[4096 + c0 * 16 + myl];
        const float fd1 = lds_raw[4096 + (c0 + 1) * 16 + myl];
        float2 f2;
        f2.x = fr0 * wr + fd0 * wd;
        f2.y = fr1 * wr + fd1 * wd;
        *(float2*)&lds_F[(cp << 5) + (myl << 1)] = f2;
      }
    }
    __syncthreads();

    // raw buffer is dead now: overlap next tile's TDM with this tile's WMMA
    if (wave == 0 && t + 1 < TPB) {
      const uint64_t pxb = (uint64_t)((tile0 + t + 1) << 4) * 4u;
      tdm_load_tile_2d(frb + pxb, raw_r_lds);
      tdm_load_tile_2d(fdb + pxb, raw_d_lds);
    }

    const int px = px0 + l;
#pragma unroll
    for (int ot = 0; ot < 2; ++ot) {
      const int o0 = ((wave << 1) + ot) << 4;
      v8f acc = {0.f, 0.f, 0.f, 0.f, 0.f, 0.f, 0.f, 0.f};
      for (int k = 0; k < CC; k += 4) {
        const int g = (k >> 1) + half;                       // c-pair group
        const v2f Ap = *(const v2f*)(Wt2 + (g << 9) + ((o0 + l) << 1));
        const float2 Bp = *(const float2*)&lds_F[(g << 5) + (l << 1)];
        v2f Bv; Bv.x = Bp.x; Bv.y = Bp.y;
        acc = __builtin_amdgcn_wmma_f32_16x16x4_f32(
            false, Ap, false, Bv, (short)0, acc, false, false);
      }
#pragma unroll
      for (int v = 0; v < 8; ++v) {
        const int o = o0 + v + (half << 3);
        out[((size_t)(b * CC + o) << 12) + px] =
            fmaxf(acc[v] * binv[ot][v] + badd[ot][v], 0.f);
      }
    }
    // next iteration's leading barrier orders F WAR and raw RAW hazards
  }
}

// ---------------------------------------------------------------------------
extern "C" void kernel_launch(void* const* d_in, const int* in_sizes, int n_in,
                              void* d_out, int out_size, void* d_ws, size_t ws_size,
                              hipStream_t stream) {
  const float* f_rgb      = (const float*)d_in[0];
  const float* f_depth    = (const float*)d_in[1];
  // d_in[2] (r_rgb) is unused by the reference
  const float* r_depth    = (const float*)d_in[3];
  const float* proto_rgb  = (const float*)d_in[4];
  const float* proto_dep  = (const float*)d_in[5];
  const float* query_mask = (const float*)d_in[6];
  const float* conv_w     = (const float*)d_in[7];
  const float* bn_gamma   = (const float*)d_in[8];
  const float* bn_beta    = (const float*)d_in[9];
  const float* bn_mean    = (const float*)d_in[10];
  const float* bn_var     = (const float*)d_in[11];

  float* ws       = (float*)d_ws;
  float* rgb_sim  = ws;                 // 32768
  float* dep_sim  = ws + 32768;         // 32768
  float* Wt2      = ws + 65536;         // 65536
  float* stats    = ws + 131072;        // 32
  float* pso_out  = ws + 131104;        // 4
  float* out      = (float*)d_out;

  sim_kernel<<<dim3(128), dim3(256), 0, stream>>>(
      f_rgb, f_depth, proto_rgb, proto_dep, rgb_sim, dep_sim);
  transpose_kernel<<<dim3(256), dim3(256), 0, stream>>>(conv_w, Wt2);
  stats_kernel<<<dim3(1), dim3(256), 0, stream>>>(
      rgb_sim, dep_sim, query_mask, r_depth, stats);
  pso_kernel<<<dim3(1), dim3(32), 0, stream>>>(stats, pso_out);
  gemm_kernel<<<dim3(64, 8), dim3(256), 0, stream>>>(
      f_rgb, f_depth, rgb_sim, dep_sim, Wt2, pso_out,
      bn_gamma, bn_beta, bn_mean, bn_var, out);
}